// DCTBlockExtractor_90598040142239
// MI455X (gfx1250) — compile-verified
//
#include <hip/hip_runtime.h>

typedef float v2f __attribute__((ext_vector_type(2)));
typedef float v8f __attribute__((ext_vector_type(8)));

// Frequency bands (u,v): (0,1),(1,0),(1,1),(2,2),(3,3),(4,4)
// A-matrix row m: band = m>>1; (m&1) ? Im : Re of e^{-2*pi*i*(u*r+v*c)/8}.
// Twiddles are only {0, +-1, +-sqrt(2)/2}; -sin(th) == cos(th + pi/2), so the
// Im part is just a phase shift of +2 (in 1/8-turn units). Branch-free select.
__device__ __forceinline__ float oct_cos(int t, float oneS, float rS) {
    // returns rowmask * cos(2*pi*t/8), t in [0,8); oneS/rS carry the row mask.
    float mag = (t & 1) ? rS : ((t & 2) ? 0.0f : oneS);
    bool neg = ((unsigned)(t - 3)) <= 2u;   // t == 3,4,5 -> negative
    return neg ? -mag : mag;
}

// One image (b,c) = 512x512 = 64x64 blocks = 4096 blocks = 256 tiles of 16 blocks.
// 32 waves per image, 8 tiles per wave. Workgroup = 256 threads = 8 waves.
__global__ __launch_bounds__(256) void dct_band_wmma_kernel(const float* __restrict__ x,
                                                            float* __restrict__ out) {
    const int lane  = threadIdx.x & 31;
    const int gwave = blockIdx.x * 8 + (threadIdx.x >> 5);
    const int image = gwave >> 5;     // (b*3 + c)
    const int wslot = gwave & 31;     // 0..31

    const float* img = x + (size_t)image * (512 * 512);

    const int off2 = (lane >> 3) & 2;   // 0 for lanes 0-15, 2 for lanes 16-31 (K sub-offset)
    const int n    = lane & 15;         // A: row M; B: block-in-tile N

    // Per-lane band parameters (rows 12..15 masked to zero via oneS/rS).
    const int band = n >> 1;
    const int im2  = (n & 1) * 2;       // phase shift: Im part = cos(theta + pi/2)
    const int u    = (band == 0) ? 0 : ((band == 1) ? 1 : (band - 1));
    const int v    = (band == 0) ? 1 : ((band == 1) ? 0 : (band - 1));
    const float rowmask = (n < 12) ? 1.0f : 0.0f;
    const float oneS = rowmask;
    const float rS   = 0.70710678118654752f * rowmask;

    // Build constant A matrix: 16 K-chunks, v2f per lane.
    // Lane holds A[M=n][e0] and A[M=n][e0+1], e0 = 4*k0 + off2; r=e>>3, c=e&7.
    v2f a[16];
#pragma unroll
    for (int k0 = 0; k0 < 16; ++k0) {
        int r = k0 >> 1;
        int c = (k0 & 1) * 4 + off2;
        int t0 = (u * r + v * c + im2) & 7;
        int t1 = (u * r + v * (c + 1) + im2) & 7;
        a[k0].x = oct_cos(t0, oneS, rS);
        a[k0].y = oct_cos(t1, oneS, rS);
    }

    float acc01 = 0.0f, acc23 = 0.0f, acc45 = 0.0f, acc67 = 0.0f;

    for (int ti = 0; ti < 8; ++ti) {
        int t  = wslot * 8 + ti;        // tile 0..255
        int tr = t >> 2;                // block-row (0..63)
        int tc = t & 3;                 // tile column (16 blocks each)

        // Lane base: element (row 0, col off2) of block (tr, tc*16+n)
        const float* base = img + (size_t)(tr * 8) * 512 + tc * 128 + n * 8 + off2;

        // Two independent accumulation chains to shorten the WMMA dep chain.
        v8f c0 = {0.f,0.f,0.f,0.f,0.f,0.f,0.f,0.f};
        v8f c1 = {0.f,0.f,0.f,0.f,0.f,0.f,0.f,0.f};
#pragma unroll
        for (int k0 = 0; k0 < 16; k0 += 2) {
            // chunk k: memory offset (k>>1)*512 floats (row) + (k&1)*4 floats (half-row)
            v2f b0 = *(const v2f*)(base + ((k0    ) >> 1) * 512 + ((k0    ) & 1) * 4);
            v2f b1 = *(const v2f*)(base + ((k0 + 1) >> 1) * 512 + ((k0 + 1) & 1) * 4);
            c0 = __builtin_amdgcn_wmma_f32_16x16x4_f32(false, a[k0],     false, b0,
                                                       (short)0, c0, false, false);
            c1 = __builtin_amdgcn_wmma_f32_16x16x4_f32(false, a[k0 + 1], false, b1,
                                                       (short)0, c1, false, false);
        }
        v8f d = c0 + c1;

        // Rows (2b, 2b+1) = (re, im) of band b. Lanes 0-15: bands 0..3 in d[0..7];
        // lanes 16-31: bands 4,5 in d[0..3] (d[4..7] are zero-padded rows).
        acc01 += __builtin_amdgcn_sqrtf(d[0] * d[0] + d[1] * d[1]);
        acc23 += __builtin_amdgcn_sqrtf(d[2] * d[2] + d[3] * d[3]);
        acc45 += __builtin_amdgcn_sqrtf(d[4] * d[4] + d[5] * d[5]);
        acc67 += __builtin_amdgcn_sqrtf(d[6] * d[6] + d[7] * d[7]);
    }

    // Reduce across each 16-lane half independently (xor 1,2,4,8).
#pragma unroll
    for (int s = 1; s < 16; s <<= 1) {
        acc01 += __shfl_xor(acc01, s, 32);
        acc23 += __shfl_xor(acc23, s, 32);
        acc45 += __shfl_xor(acc45, s, 32);
        acc67 += __shfl_xor(acc67, s, 32);
    }

    const float scale = 1.0f / 4096.0f;   // mean over 64x64 blocks
    float* o = out + image * 6;
    if (lane == 0) {
        atomicAdd(o + 0, acc01 * scale);  // band (0,1)
        atomicAdd(o + 1, acc23 * scale);  // band (1,0)
        atomicAdd(o + 2, acc45 * scale);  // band (1,1)
        atomicAdd(o + 3, acc67 * scale);  // band (2,2)
    } else if (lane == 16) {
        atomicAdd(o + 4, acc01 * scale);  // band (3,3)
        atomicAdd(o + 5, acc23 * scale);  // band (4,4)
    }
}

extern "C" void kernel_launch(void* const* d_in, const int* in_sizes, int n_in,
                              void* d_out, int out_size, void* d_ws, size_t ws_size,
                              hipStream_t stream) {
    const float* x = (const float*)d_in[0];
    float* out = (float*)d_out;

    // Output is accumulated via atomics; zero it first (capture-safe memset node).
    hipMemsetAsync(d_out, 0, (size_t)out_size * sizeof(float), stream);

    const int images = in_sizes[0] >> 18;        // elements / (512*512) = B*C
    const int wgs = images * 4;                  // 8 waves/WG, 32 waves per image
    dct_band_wmma_kernel<<<wgs, 256, 0, stream>>>(x, out);
}